// DynamicMinkowskiConvolution_78632261255714
// MI455X (gfx1250) — compile-verified
//
#include <hip/hip_runtime.h>

typedef __attribute__((ext_vector_type(2))) float v2f;
typedef __attribute__((ext_vector_type(8))) float v8f;

// Problem constants (fixed by the reference).
constexpr int NVOX  = 100000;
constexpr int KOFF  = 27;
constexpr int LPAIR = 50000;
constexpr int INC   = 64;
constexpr int OUTC  = 64;

constexpr int M_BLOCK     = 128;       // gathered rows per block (4 M-tiles per wave)
constexpr int FEAT_STRIDE = INC  + 4;  // 68 floats: bank-conflict padding
constexpr int KER_STRIDE  = OUTC + 4;  // 68 floats

__global__ void zero_out_kernel(float* __restrict__ out, int n) {
    int i = blockIdx.x * blockDim.x + threadIdx.x;
    if (i < n) out[i] = 0.0f;
}

// CDNA5 async global->LDS copy, 16B per lane, immediate offset folded into the
// instruction (INST_OFFSET is added to BOTH the LDS and the global address).
template <int OFF>
__device__ __forceinline__ void async_ld_b128(unsigned lds_off, const void* gaddr) {
    asm volatile("global_load_async_to_lds_b128 %0, %1, off offset:%c2"
                 :: "v"(lds_off), "v"(gaddr), "i"(OFF)
                 : "memory");
}
__device__ __forceinline__ void wait_async0() {
    asm volatile("s_wait_asynccnt 0x0" ::: "memory");
}
// Generic pointer to a __shared__ object -> wave-relative LDS byte offset
// (flat LDS aperture keeps the allocation offset in the low 32 bits).
__device__ __forceinline__ unsigned lds_byte_off(const void* p) {
    return (unsigned)(unsigned long long)p;
}

__global__ __launch_bounds__(256)
void spconv_wmma_f32(const float* __restrict__ features,   // [NVOX, INC]
                     const float* __restrict__ kern,       // [KOFF, INC, OUTC]
                     const int*   __restrict__ nbr_in,     // [KOFF, LPAIR]
                     const int*   __restrict__ nbr_out,    // [KOFF, LPAIR]
                     float*       __restrict__ out) {      // [NVOX, OUTC]
    __shared__ float feat[M_BLOCK * FEAT_STRIDE];  // gathered A rows (34.8 KB)
    __shared__ float ker [INC     * KER_STRIDE];   // B = kernel[k]   (17.4 KB)
    __shared__ int   rowout[M_BLOCK];              // scatter dests (-1 = invalid)

    const int k    = blockIdx.y;
    const int base = blockIdx.x * M_BLOCK;
    const int tid  = threadIdx.x;

    // ---- Stage B: kernel[k] 64x64 -> padded LDS via async DMA ----
    {
        const int r  = tid >> 2;          // 0..63  input channel
        const int c0 = (tid & 3) * 16;    // 16-float chunk
        const float* kp = kern + (size_t)k * INC * OUTC + r * OUTC + c0;
        const unsigned lk = lds_byte_off(&ker[r * KER_STRIDE + c0]);
        async_ld_b128<0>(lk, kp);
        async_ld_b128<16>(lk, kp);
        async_ld_b128<32>(lk, kp);
        async_ld_b128<48>(lk, kp);
    }

    // ---- Stage A: gather 128 feature rows via nbr_in (async DMA) ----
    {
        if (tid < M_BLOCK) {
            const int gr = base + tid;
            rowout[tid] = (gr < LPAIR) ? nbr_out[(size_t)k * LPAIR + gr] : -1;
        }
        const int r  = tid >> 1;          // 0..127 gathered row
        const int c0 = (tid & 1) * 32;    // 32-float half-row
        const int gr = base + r;
        // Tail rows: clamp to row 0 (data unused: their scatter rows are -1).
        const int src = (gr < LPAIR) ? nbr_in[(size_t)k * LPAIR + gr] : 0;
        const float* fp = features + (size_t)src * INC + c0;
        const unsigned lf = lds_byte_off(&feat[r * FEAT_STRIDE + c0]);
        async_ld_b128<0>(lf, fp);
        async_ld_b128<16>(lf, fp);
        async_ld_b128<32>(lf, fp);
        async_ld_b128<48>(lf, fp);
        async_ld_b128<64>(lf, fp);
        async_ld_b128<80>(lf, fp);
        async_ld_b128<96>(lf, fp);
        async_ld_b128<112>(lf, fp);
    }

    wait_async0();      // this wave's DMA has landed in LDS
    __syncthreads();    // ...and is visible to every wave in the block

    // ---- WMMA compute: wave -> N-tile nt, M-tiles {mtb, mtb+2, mtb+4, mtb+6} ----
    const int wave = tid >> 5;          // 0..7
    const int lane = tid & 31;
    const int mtb  = wave >> 2;         // 0..1
    const int nt   = wave & 3;          // 0..3
    const int ln   = lane & 15;
    const int hi   = lane >> 4;         // 0: K{0,1}, 1: K{2,3} halves

    int arow[4];
    #pragma unroll
    for (int t = 0; t < 4; ++t) arow[t] = (mtb + 2 * t) * 16 + ln;
    const int ncol = nt * 16 + ln;      // B/D column (N = lane&15)

    v8f acc[4] = {};

    #pragma unroll
    for (int kk = 0; kk < 16; ++kk) {
        const int kb = kk * 4 + hi * 2;     // this lane's K base within the 4-slice
        // B fragment (shared by 4 WMMAs): b.x = B[kb][N], b.y = B[kb+1][N]
        v2f b;
        b.x = ker[(kb)     * KER_STRIDE + ncol];
        b.y = ker[(kb + 1) * KER_STRIDE + ncol];
        #pragma unroll
        for (int t = 0; t < 4; ++t) {
            // A fragment: a.x = A[M][kb], a.y = A[M][kb+1] (contiguous, 8B aligned)
            v2f a = *(const v2f*)(&feat[arow[t] * FEAT_STRIDE + kb]);
            acc[t] = __builtin_amdgcn_wmma_f32_16x16x4_f32(
                /*neg_a=*/false, a, /*neg_b=*/false, b,
                /*c_mod=*/(short)0, acc[t], /*reuse_a=*/false, /*reuse_b=*/false);
        }
    }

    // ---- Scatter-add: D layout VGPR v -> M = v + 8*hi, N = lane&15 ----
    #pragma unroll
    for (int t = 0; t < 4; ++t) {
        #pragma unroll
        for (int v = 0; v < 8; ++v) {
            const int lr = (mtb + 2 * t) * 16 + v + hi * 8;
            const int d  = rowout[lr];
            if (d >= 0) unsafeAtomicAdd(&out[(size_t)d * OUTC + ncol], acc[t][v]);
        }
    }
}

extern "C" void kernel_launch(void* const* d_in, const int* in_sizes, int n_in,
                              void* d_out, int out_size, void* d_ws, size_t ws_size,
                              hipStream_t stream) {
    const float* features = (const float*)d_in[0];   // [NVOX, INC] f32
    const float* kern     = (const float*)d_in[1];   // [KOFF, INC, OUTC] f32
    const int*   nbr_in   = (const int*)d_in[2];     // [KOFF, LPAIR] i32
    const int*   nbr_out  = (const int*)d_in[3];     // [KOFF, LPAIR] i32
    float*       out      = (float*)d_out;           // [NVOX, OUTC] f32

    const int total = NVOX * OUTC;
    zero_out_kernel<<<(total + 255) / 256, 256, 0, stream>>>(out, total);

    dim3 grid((LPAIR + M_BLOCK - 1) / M_BLOCK, KOFF);
    spconv_wmma_f32<<<grid, 256, 0, stream>>>(features, kern, nbr_in, nbr_out, out);
}